// PrototypeMemory_33638183862566
// MI455X (gfx1250) — compile-verified
//
#include <hip/hip_runtime.h>
#include <math.h>

#define DDIM   2048
#define KPROTO 16384
#define THRESH 0.5f

typedef float v2f __attribute__((ext_vector_type(2)));
typedef float v8f __attribute__((ext_vector_type(8)));

// D = A(16x4, f32) * B(4x16, f32) + C(16x16, f32)
// A layout: lanes 0-15 -> M=lane, v0=K0, v1=K1 ; lanes 16-31 -> M=lane-16, v0=K2, v1=K3
// B is z replicated across all 16 columns, so every column of D = dot(P_row, z_chunk).
__device__ __forceinline__ v8f wmma_dot4(v2f a, v2f b, v8f c) {
  return __builtin_amdgcn_wmma_f32_16x16x4_f32(
      /*neg_a=*/false, a, /*neg_b=*/false, b,
      /*c_mod=*/(short)0, c, /*reuse_a=*/false, /*reuse_b=*/false);
}

// ---------------------------------------------------------------------------
// Kernel A: streaming pass over the 128 MB prototype bank.
// Each wave owns a 16-row tile; computes dist0[k] = ||P_k||^2 - 2*P_k.z
// (the constant ||z||^2 is added in the finalize kernel).
// ---------------------------------------------------------------------------
__global__ __launch_bounds__(256) void proto_score_kernel(
    const float* __restrict__ protos, const float* __restrict__ z,
    float* __restrict__ dist0) {
  __shared__ float lds_dot[8][16];
  __shared__ float lds_sq[8][32];

  const int tid  = threadIdx.x;
  const int wv   = tid >> 5;          // wave id within block (8 waves)
  const int lane = tid & 31;          // wave32
  const int r    = lane & 15;         // prototype row within tile
  const int half = lane >> 4;         // 0: K pair {0,1}, 1: K pair {2,3}
  const int tile = blockIdx.x * 8 + wv;            // 1024 tiles of 16 rows
  const int row  = tile * 16 + r;

  // lane r streams P[row][8t .. 8t+3], lane r+16 streams P[row][8t+4 .. 8t+7]
  const float* prow = protos + (size_t)row * DDIM + half * 4;
  const float* zrow = z + half * 4;

  v8f c = {0.f, 0.f, 0.f, 0.f, 0.f, 0.f, 0.f, 0.f};
  float sq = 0.f;

#pragma unroll 4
  for (int t = 0; t < DDIM / 8; ++t) {
    const float4 q  = *(const float4*)(prow + t * 8);
    const float4 zq = *(const float4*)(zrow + t * 8);
    // WMMA #1 covers physical k = {8t,8t+1} (lanes 0-15) / {8t+4,8t+5} (lanes 16-31)
    v2f a1 = {q.x, q.y}; v2f b1 = {zq.x, zq.y};
    // WMMA #2 covers physical k = {8t+2,8t+3} / {8t+6,8t+7}
    v2f a2 = {q.z, q.w}; v2f b2 = {zq.z, zq.w};
    c = wmma_dot4(a1, b1, c);
    c = wmma_dot4(a2, b2, c);
    // ||P_row||^2 partial from the same registers (no extra traffic)
    sq = fmaf(q.x, q.x, sq);
    sq = fmaf(q.y, q.y, sq);
    sq = fmaf(q.z, q.z, sq);
    sq = fmaf(q.w, q.w, sq);
  }

  // D layout: lane 0 col 0 holds rows 0..7 in c[0..7]; lane 16 col 0 holds rows 8..15.
  lds_sq[wv][lane] = sq;
  if (lane == 0) {
#pragma unroll
    for (int i = 0; i < 8; ++i) lds_dot[wv][i] = c[i];
  }
  if (lane == 16) {
#pragma unroll
    for (int i = 0; i < 8; ++i) lds_dot[wv][8 + i] = c[i];
  }
  __syncthreads();

  if (lane < 16) {
    float dd = lds_sq[wv][lane] + lds_sq[wv][lane + 16]
             - 2.0f * lds_dot[wv][lane];
    dist0[tile * 16 + lane] = dd;   // coalesced 16-float store per wave
  }
}

// ---------------------------------------------------------------------------
// Block-wide reductions (1024 threads = 32 waves of wave32)
// ---------------------------------------------------------------------------
__device__ __forceinline__ float block_sum(float v, float* lds) {
#pragma unroll
  for (int o = 16; o > 0; o >>= 1) v += __shfl_down(v, o, 32);
  const int wv = threadIdx.x >> 5, lane = threadIdx.x & 31;
  if (lane == 0) lds[wv] = v;
  __syncthreads();
  if (wv == 0) {
    v = lds[lane];
#pragma unroll
    for (int o = 16; o > 0; o >>= 1) v += __shfl_down(v, o, 32);
    if (lane == 0) lds[0] = v;
  }
  __syncthreads();
  v = lds[0];
  __syncthreads();
  return v;
}

// min with first-index tie-break (matches jnp.argmin / argmax-of-negated)
__device__ __forceinline__ void block_min_idx(float& v, int& idx,
                                              float* ldsv, int* ldsi) {
#pragma unroll
  for (int o = 16; o > 0; o >>= 1) {
    float ov = __shfl_down(v, o, 32);
    int   oi = __shfl_down(idx, o, 32);
    if (ov < v || (ov == v && oi < idx)) { v = ov; idx = oi; }
  }
  const int wv = threadIdx.x >> 5, lane = threadIdx.x & 31;
  if (lane == 0) { ldsv[wv] = v; ldsi[wv] = idx; }
  __syncthreads();
  if (wv == 0) {
    v = ldsv[lane]; idx = ldsi[lane];
#pragma unroll
    for (int o = 16; o > 0; o >>= 1) {
      float ov = __shfl_down(v, o, 32);
      int   oi = __shfl_down(idx, o, 32);
      if (ov < v || (ov == v && oi < idx)) { v = ov; idx = oi; }
    }
    if (lane == 0) { ldsv[0] = v; ldsi[0] = idx; }
  }
  __syncthreads();
  v = ldsv[0]; idx = ldsi[0];
  __syncthreads();
}

// ---------------------------------------------------------------------------
// Kernel B: all K-sized reductions + gate + rank-1 re-score (no 2nd HBM pass).
// dist_new differs from dist0 at exactly one row:
//   take_new : dist_new[idx_new] = 0
//   match    : dist_new[idx_ema] = wp^2 * (dist0[idx_ema] + ||z||^2)
// ---------------------------------------------------------------------------
__global__ __launch_bounds__(1024) void proto_finalize_kernel(
    const float* __restrict__ z, const float* __restrict__ usages,
    const float* __restrict__ beta, const float* __restrict__ gamma,
    const float* __restrict__ dist0, float* __restrict__ out) {
  __shared__ float ldsv[32];
  __shared__ int   ldsi[32];
  __shared__ float s_zz, s_u, s_dnew;
  __shared__ int   s_idx;

  const int tid = threadIdx.x;

  // ---- ||z||^2 ----
  float s = 0.f;
  for (int i = tid; i < DDIM; i += 1024) { float zi = z[i]; s = fmaf(zi, zi, s); }
  const float zz = block_sum(s, ldsv);

  // ---- min dist0 + argmin (= idx_ema, since softmax(-d) argmax == argmin d) ----
  float mv = 3.402823466e38f; int mi = KPROTO - 1;
  for (int i = tid; i < KPROTO; i += 1024) {
    float d = dist0[i];
    if (d < mv) { mv = d; mi = i; }
  }
  block_min_idx(mv, mi, ldsv, ldsi);

  // ---- argmin usages (= idx_new) ----
  float uv = 3.402823466e38f; int ui = KPROTO - 1;
  for (int i = tid; i < KPROTO; i += 1024) {
    float uu = usages[i];
    if (uu < uv) { uv = uu; ui = i; }
  }
  block_min_idx(uv, ui, ldsv, ldsi);

  // ---- softmax max value: y_max = 1 / sum_k exp(dmin - d_k) (zz cancels) ----
  float s1 = 0.f;
  for (int i = tid; i < KPROTO; i += 1024) s1 += expf(mv - dist0[i]);
  const float S1 = block_sum(s1, ldsv);

  if (tid == 0) {
    const float dfull_min = mv + zz;
    const float uval = 1.0f / (1.0f + expf(-((dfull_min - beta[0]) / gamma[0])));
    const int take_new = (uval >= THRESH) ? 1 : 0;
    const float max_y = 1.0f / S1;
    const float delta = max_y * (1.0f - uval);
    const float use   = usages[mi];
    const float wp    = use / (use + delta);     // new_row - z = wp*(P[idx]-z)
    s_idx  = take_new ? ui : mi;
    s_dnew = take_new ? 0.0f : wp * wp * dfull_min;
    s_u    = uval;
    s_zz   = zz;
  }
  __syncthreads();
  const float zzb  = s_zz;
  const int   idx  = s_idx;
  const float dnew = s_dnew;

  // ---- label = argmin over patched full distances ----
  float m2 = 3.402823466e38f; int l2 = KPROTO - 1;
  for (int i = tid; i < KPROTO; i += 1024) {
    float d = (i == idx) ? dnew : (dist0[i] + zzb);
    if (d < m2) { m2 = d; l2 = i; }
  }
  block_min_idx(m2, l2, ldsv, ldsi);

  // ---- loss = log sum_k exp(d_label - d_k)  (label is the min -> stable) ----
  float s2 = 0.f;
  for (int i = tid; i < KPROTO; i += 1024) {
    float d = (i == idx) ? dnew : (dist0[i] + zzb);
    s2 += expf(m2 - d);
  }
  const float S2 = block_sum(s2, ldsv);

  if (tid == 0) {
    out[0] = logf(S2);      // loss
    out[1] = (float)l2;     // label
    out[2] = s_u;           // u
  }
}

// ---------------------------------------------------------------------------
extern "C" void kernel_launch(void* const* d_in, const int* in_sizes, int n_in,
                              void* d_out, int out_size, void* d_ws, size_t ws_size,
                              hipStream_t stream) {
  const float* z      = (const float*)d_in[0];  // [1, 2048]
  const float* protos = (const float*)d_in[1];  // [16384, 2048]
  const float* usages = (const float*)d_in[2];  // [16384]
  const float* beta   = (const float*)d_in[3];  // [1]
  const float* gamma  = (const float*)d_in[4];  // [1]
  float* dist0 = (float*)d_ws;                  // K floats = 64 KB scratch

  // 1024 row-tiles of 16, 8 waves per 256-thread block -> 128 blocks
  proto_score_kernel<<<128, 256, 0, stream>>>(protos, z, dist0);
  proto_finalize_kernel<<<1, 1024, 0, stream>>>(z, usages, beta, gamma, dist0,
                                                (float*)d_out);
}